// GNNML1_64991445123447
// MI455X (gfx1250) — compile-verified
//
#include <hip/hip_runtime.h>

typedef float v2f __attribute__((ext_vector_type(2)));
typedef float v8f __attribute__((ext_vector_type(8)));

#define NOUT 32

// ---------------------------------------------------------------------------
// Fused block GEMM: per node row i (16-row tiles, one wave per tile):
//   hc[i,:]      = h[i,:] @ Wc                      (scatter input, bias later)
//   partial[i,:] = relu(h@Wa + ba) + relu((h@Wm1 + bm1) * (h@Wm2 + bm2))
// using full-precision V_WMMA_F32_16X16X4_F32 (K-step = 4).
//
// Weights are staged in LDS PRE-SWIZZLED into B-fragment order: for each
// (matrix, k-step, col-half, lane) the lane's two B values (rows kr, kr+1 with
// kr = k0 + 2*(lane>=16)) are contiguous, so the inner loop needs exactly one
// aligned ds_load_b64 per WMMA that lands directly in an even VGPR pair.
// ---------------------------------------------------------------------------
template <int F>
__global__ __launch_bounds__(256) void block_gemm_kernel(
    const float* __restrict__ h,                       // [N, F]
    const float* __restrict__ Wa,  const float* __restrict__ ba,
    const float* __restrict__ Wm1, const float* __restrict__ bm1,
    const float* __restrict__ Wm2, const float* __restrict__ bm2,
    const float* __restrict__ Wc,                      // [F, 32] each
    float* __restrict__ hc_out,                        // [N, 32]
    float* __restrict__ partial_out,                   // [N, 32]
    int nTiles)
{
    constexpr int KSTEPS = F / 4;
    constexpr int NPAIRS = 4 * KSTEPS * 2 * 32;        // fragment pairs total
    __shared__ float sWf[2 * NPAIRS];                  // == 4*F*32 floats
    __shared__ float sB[3 * NOUT];

    const float* Wsrc[4] = {Wa, Wm1, Wm2, Wc};
    const int tid = threadIdx.x;

    // Swizzled preload: p = ((mat*KSTEPS + s)*2 + ch)*32 + l
    for (int p = tid; p < NPAIRS; p += 256) {
        const int l   = p & 31;
        const int q   = p >> 5;
        const int ch  = q & 1;
        const int r   = q >> 1;
        const int s   = r & (KSTEPS - 1);              // KSTEPS is power of 2
        const int mat = r / KSTEPS;
        const int kr  = 4 * s + ((l >> 4) << 1);
        const int col = ch * 16 + (l & 15);
        sWf[2 * p + 0] = Wsrc[mat][kr * NOUT + col];
        sWf[2 * p + 1] = Wsrc[mat][(kr + 1) * NOUT + col];
    }
    if (tid < NOUT) {
        sB[tid]            = ba[tid];
        sB[NOUT + tid]     = bm1[tid];
        sB[2 * NOUT + tid] = bm2[tid];
    }
    __syncthreads();

    const int wave = tid >> 5;
    const int lane = tid & 31;
    const int tile = blockIdx.x * 8 + wave;
    if (tile >= nTiles) return;   // wave-uniform: EXEC stays all-ones for WMMA

    const int m0    = tile * 16;
    const int nlane = lane & 15;          // N index for B/C/D, M row for A
    const int khalf = (lane >> 4) << 1;   // 0 (lanes 0-15) or 2 (lanes 16-31)

    v8f acc[8] = {};                      // [Wa lo/hi, Wm1 lo/hi, Wm2 lo/hi, Wc lo/hi]

    const float* hrow = h + (size_t)(m0 + nlane) * F;

    for (int s = 0; s < KSTEPS; ++s) {
        // A fragment 16x4 f32: lanes 0-15 rows M with K=k0,k0+1; lanes 16-31 K=k0+2,k0+3
        v2f a = *(const v2f*)(hrow + 4 * s + khalf);
#pragma unroll
        for (int t = 0; t < 8; ++t) {
            const int mat = t >> 1;
            const int ch  = t & 1;
            // Single aligned b64 LDS read -> even VGPR pair
            v2f b = *(const v2f*)&sWf[((((mat * KSTEPS + s) * 2 + ch) * 32) + lane) * 2];
            acc[t] = __builtin_amdgcn_wmma_f32_16x16x4_f32(
                false, a, false, b, (short)0, acc[t], false, false);
        }
    }

    // C/D layout: VGPR v -> row m0+v (lanes 0-15) / m0+v+8 (lanes 16-31), col = lane&15
    const int rowoff = (lane >> 4) * 8;
#pragma unroll
    for (int v = 0; v < 8; ++v) {
        const int m = m0 + rowoff + v;
#pragma unroll
        for (int ch = 0; ch < 2; ++ch) {
            const int col = ch * 16 + nlane;
            const float va  = acc[0 + ch][v] + sB[col];
            const float vm1 = acc[2 + ch][v] + sB[NOUT + col];
            const float vm2 = acc[4 + ch][v] + sB[2 * NOUT + col];
            const float vc  = acc[6 + ch][v];
            const size_t o = (size_t)m * NOUT + col;
            hc_out[o]      = vc;
            partial_out[o] = fmaxf(va, 0.f) + fmaxf(vm1 * vm2, 0.f);
        }
    }
}

// ---------------------------------------------------------------------------
// agg = 0
// ---------------------------------------------------------------------------
__global__ void zero_kernel(float4* __restrict__ p, int n4) {
    int i = blockIdx.x * blockDim.x + threadIdx.x;
    if (i < n4) p[i] = make_float4(0.f, 0.f, 0.f, 0.f);
}

// ---------------------------------------------------------------------------
// agg[dst] += hc[src]; one lane per (edge, feature) so each wave issues 32
// consecutive gathers + 32 consecutive f32 atomics (L2-resident working set).
// ---------------------------------------------------------------------------
__global__ void scatter_kernel(const int* __restrict__ ei,     // [2, nE]
                               const float* __restrict__ hc,   // [N, 32]
                               float* __restrict__ agg,        // [N, 32]
                               int nE)
{
    long long gid = (long long)blockIdx.x * blockDim.x + threadIdx.x;
    long long e = gid >> 5;
    if (e >= nE) return;
    int f = (int)(gid & 31);
    int s = ei[e];
    int d = ei[(long long)nE + e];
    atomicAdd(agg + (size_t)d * NOUT + f, hc[(size_t)s * NOUT + f]);
}

// ---------------------------------------------------------------------------
// h_next = partial + relu(agg + bc)   (in place over `partial`)
// ---------------------------------------------------------------------------
__global__ void finalize_kernel(float* __restrict__ hnext,
                                const float* __restrict__ agg,
                                const float* __restrict__ bc,
                                long long total)
{
    long long i = (long long)blockIdx.x * blockDim.x + threadIdx.x;
    if (i >= total) return;
    hnext[i] = hnext[i] + fmaxf(agg[i] + bc[(int)(i & (NOUT - 1))], 0.f);
}

// ---------------------------------------------------------------------------
// out = h @ W2 + b2      ([N,32] x [32,1])
// ---------------------------------------------------------------------------
__global__ void out_kernel(const float* __restrict__ h,
                           const float* __restrict__ W2,
                           const float* __restrict__ b2,
                           float* __restrict__ out, int N)
{
    int i = blockIdx.x * blockDim.x + threadIdx.x;
    if (i >= N) return;
    const float4* hv = (const float4*)(h + (size_t)i * NOUT);
    float s = 0.f;
#pragma unroll
    for (int c = 0; c < 8; ++c) {
        float4 hh = hv[c];
        s += hh.x * W2[c * 4 + 0] + hh.y * W2[c * 4 + 1] +
             hh.z * W2[c * 4 + 2] + hh.w * W2[c * 4 + 3];
    }
    out[i] = s + b2[0];
}

// ---------------------------------------------------------------------------
// Launcher
// ---------------------------------------------------------------------------
extern "C" void kernel_launch(void* const* d_in, const int* in_sizes, int n_in,
                              void* d_out, int out_size, void* d_ws, size_t ws_size,
                              hipStream_t stream)
{
    const float* x   = (const float*)d_in[0];
    const int*   ei  = (const int*)d_in[1];
    const float* Wc1 = (const float*)d_in[2];  const float* bc1 = (const float*)d_in[3];
    const float* Wc2 = (const float*)d_in[4];  const float* bc2 = (const float*)d_in[5];
    const float* Wc3 = (const float*)d_in[6];  const float* bc3 = (const float*)d_in[7];
    const float* W11 = (const float*)d_in[8];  const float* b11 = (const float*)d_in[9];
    const float* W12 = (const float*)d_in[10]; const float* b12 = (const float*)d_in[11];
    const float* W13 = (const float*)d_in[12]; const float* b13 = (const float*)d_in[13];
    const float* W21 = (const float*)d_in[14]; const float* b21 = (const float*)d_in[15];
    const float* W22 = (const float*)d_in[16]; const float* b22 = (const float*)d_in[17];
    const float* W23 = (const float*)d_in[18]; const float* b23 = (const float*)d_in[19];
    const float* W31 = (const float*)d_in[20]; const float* b31 = (const float*)d_in[21];
    const float* W32 = (const float*)d_in[22]; const float* b32 = (const float*)d_in[23];
    const float* W33 = (const float*)d_in[24]; const float* b33 = (const float*)d_in[25];
    const float* W2  = (const float*)d_in[26]; const float* b2  = (const float*)d_in[27];

    const int N  = in_sizes[0] / 128;   // 100000
    const int nE = in_sizes[1] / 2;     // 3200000

    const size_t stride = (size_t)N * NOUT;      // floats per [N,32] buffer
    float* ws   = (float*)d_ws;
    float* bufA = ws;                             // h after block 1 / block 3
    float* bufB = ws + stride;                    // h after block 2
    float* hc   = ws + 2 * stride;                // h @ Wc (scatter source)
    float* agg  = ws + 3 * stride;                // scatter accumulator

    const int nTiles = (N + 15) / 16;
    dim3 gemmGrid((nTiles + 7) / 8);
    const int n4      = (int)(stride / 4);
    const int zBlocks = (n4 + 255) / 256;
    const long long scatWork = (long long)nE * NOUT;
    const int sBlocks = (int)((scatWork + 255) / 256);
    const long long tot = (long long)stride;
    const int fBlocks = (int)((tot + 255) / 256);

    // ---- Block 1 (F = 128) ----
    block_gemm_kernel<128><<<gemmGrid, 256, 0, stream>>>(
        x, W11, b11, W12, b12, W13, b13, Wc1, hc, bufA, nTiles);
    zero_kernel<<<zBlocks, 256, 0, stream>>>((float4*)agg, n4);
    scatter_kernel<<<sBlocks, 256, 0, stream>>>(ei, hc, agg, nE);
    finalize_kernel<<<fBlocks, 256, 0, stream>>>(bufA, agg, bc1, tot);

    // ---- Block 2 (F = 32) ----
    block_gemm_kernel<32><<<gemmGrid, 256, 0, stream>>>(
        bufA, W21, b21, W22, b22, W23, b23, Wc2, hc, bufB, nTiles);
    zero_kernel<<<zBlocks, 256, 0, stream>>>((float4*)agg, n4);
    scatter_kernel<<<sBlocks, 256, 0, stream>>>(ei, hc, agg, nE);
    finalize_kernel<<<fBlocks, 256, 0, stream>>>(bufB, agg, bc2, tot);

    // ---- Block 3 (F = 32) ----
    block_gemm_kernel<32><<<gemmGrid, 256, 0, stream>>>(
        bufB, W31, b31, W32, b32, W33, b33, Wc3, hc, bufA, nTiles);
    zero_kernel<<<zBlocks, 256, 0, stream>>>((float4*)agg, n4);
    scatter_kernel<<<sBlocks, 256, 0, stream>>>(ei, hc, agg, nE);
    finalize_kernel<<<fBlocks, 256, 0, stream>>>(bufA, agg, bc3, tot);

    // ---- Head ----
    out_kernel<<<(N + 255) / 256, 256, 0, stream>>>(bufA, W2, b2, (float*)d_out, N);
}